// Net_GCN_11227044511897
// MI455X (gfx1250) — compile-verified
//
#include <hip/hip_runtime.h>
#include <hip/hip_bf16.h>
#include <math.h>

typedef float v2f __attribute__((ext_vector_type(2)));
typedef float v8f __attribute__((ext_vector_type(8)));

// ---------------------------------------------------------------------------
// Small utility kernels
// ---------------------------------------------------------------------------
__global__ void k_fill(float* __restrict__ p, float v, int n) {
    int t = blockIdx.x * blockDim.x + threadIdx.x;
    if (t < n) p[t] = v;
}

__global__ void k_deg(const int* __restrict__ dst, float* __restrict__ deg, int E) {
    int e = blockIdx.x * blockDim.x + threadIdx.x;
    if (e < E) atomicAdd(&deg[dst[e]], 1.0f);
}

__global__ void k_rsqrt_inplace(float* __restrict__ p, int n) {
    int t = blockIdx.x * blockDim.x + threadIdx.x;
    if (t < n) p[t] = rsqrtf(p[t]);
}

// ---------------------------------------------------------------------------
// GEMM: Out[rows,64] = A[rows,64] @ W[64,64], fp32 WMMA (16x16x4), one wave
// per 16x16 output tile. rows must be a multiple of 16 (N=50000 = 3125*16).
// A-frag layout (16x4 f32): lane&15 = M row, (lane>>4)*2 = K half select.
// B-frag layout (4x16 f32): lane&15 = N col, (lane>>4)*2 = K half select.
// C/D layout (16x16 f32, 8 VGPRs): vgpr j -> M = j + 8*(lane>>4), N = lane&15.
// ---------------------------------------------------------------------------
__global__ __launch_bounds__(256) void k_gemm64_wmma(
    const float* __restrict__ A, const float* __restrict__ W,
    float* __restrict__ Out, int totalTiles) {
    int wave = (blockIdx.x * blockDim.x + threadIdx.x) >> 5;
    if (wave >= totalTiles) return;           // uniform per wave: EXEC stays all-1
    int lane = threadIdx.x & 31;
    int mt = wave >> 2;                       // row tile (rows/16 of them)
    int nt = wave & 3;                        // col tile (64/16 = 4 of them)
    int m0 = mt * 16;
    int n0 = nt * 16;
    int r = lane & 15;
    int khalf = (lane >> 4) << 1;             // 0 or 2

    const float* arow = A + (size_t)(m0 + r) * 64;
    const float* bcol = W + n0 + r;

    v8f acc = {};
#pragma unroll
    for (int k = 0; k < 64; k += 4) {
        v2f a, b;
        a.x = arow[k + khalf];
        a.y = arow[k + khalf + 1];
        b.x = bcol[(k + khalf) * 64];
        b.y = bcol[(k + khalf + 1) * 64];
        acc = __builtin_amdgcn_wmma_f32_16x16x4_f32(
            /*neg_a=*/false, a, /*neg_b=*/false, b,
            /*c_mod=*/(short)0, acc, /*reuse_a=*/false, /*reuse_b=*/false);
    }

    int row = m0 + ((lane >> 4) << 3);        // +8 for upper half lanes
    int col = n0 + r;
#pragma unroll
    for (int j = 0; j < 8; ++j)
        Out[(size_t)(row + j) * 64 + col] = acc[j];
}

// ---------------------------------------------------------------------------
// Edge scatter: agg[dst] += tmp[src] * (dis[src]*dis[dst]).
// 16 threads per edge, each handles 4 consecutive features (float4 gather,
// 4 f32 atomics). All h data is L2-resident (12.8MB << 192MB L2).
// ---------------------------------------------------------------------------
__global__ __launch_bounds__(256) void k_scatter(
    const int* __restrict__ src, const int* __restrict__ dst,
    const float* __restrict__ dis, const float* __restrict__ tmp,
    float* __restrict__ agg, int E) {
    int t = blockIdx.x * blockDim.x + threadIdx.x;
    int e = t >> 4;
    if (e >= E) return;
    int q = (t & 15) << 2;
    int s = src[e];
    int d = dst[e];
    float w = dis[s] * dis[d];
    const float4 v = *(const float4*)(tmp + (size_t)s * 64 + q);
    float* o = agg + (size_t)d * 64 + q;
    atomicAdd(o + 0, v.x * w);
    atomicAdd(o + 1, v.y * w);
    atomicAdd(o + 2, v.z * w);
    atomicAdd(o + 3, v.w * w);
}

// Self-loop term + bias + ReLU, written in place over agg.
__global__ void k_finalize(float* __restrict__ agg, const float* __restrict__ tmp,
                           const float* __restrict__ dis, const float* __restrict__ bias,
                           int total) {
    int t = blockIdx.x * blockDim.x + threadIdx.x;
    if (t >= total) return;
    int node = t >> 6;
    int f = t & 63;
    float dv = dis[node];
    float v = agg[t] + tmp[t] * (dv * dv) + bias[f];
    agg[t] = v > 0.0f ? v : 0.0f;
}

// ---------------------------------------------------------------------------
// Global mean pool + heads
// ---------------------------------------------------------------------------
__global__ void k_pool_cnt(const int* __restrict__ batch, float* __restrict__ cnt, int n) {
    int t = blockIdx.x * blockDim.x + threadIdx.x;
    if (t < n) atomicAdd(&cnt[batch[t]], 1.0f);
}

__global__ void k_pool_sum(const float* __restrict__ h, const int* __restrict__ batch,
                           float* __restrict__ sums, int total) {
    int t = blockIdx.x * blockDim.x + threadIdx.x;
    if (t >= total) return;
    int node = t >> 6;
    int f = t & 63;
    atomicAdd(&sums[(size_t)batch[node] * 64 + f], h[t]);
}

// hid1[g,j] = relu( (sums[g,:]/max(cnt,1)) @ W[:,j] + b[j] )
__global__ void k_lin1(const float* __restrict__ sums, const float* __restrict__ cnt,
                       const float* __restrict__ W, const float* __restrict__ b,
                       float* __restrict__ out, int total) {
    int t = blockIdx.x * blockDim.x + threadIdx.x;
    if (t >= total) return;
    int g = t >> 6;
    int j = t & 63;
    float c = cnt[g];
    float inv = 1.0f / (c > 1.0f ? c : 1.0f);
    float acc = b[j];
    const float* row = sums + (size_t)g * 64;
#pragma unroll 8
    for (int k = 0; k < 64; ++k) acc = fmaf(row[k] * inv, W[k * 64 + j], acc);
    out[t] = acc > 0.0f ? acc : 0.0f;
}

// logits = hid1 @ W2 + b2 ; out = log_softmax(logits). One thread per graph.
__global__ void k_lin2_logsoftmax(const float* __restrict__ h, const float* __restrict__ W,
                                  const float* __restrict__ b, float* __restrict__ out, int G) {
    int g = blockIdx.x * blockDim.x + threadIdx.x;
    if (g >= G) return;
    const float* row = h + (size_t)g * 64;
    float logits[10];
#pragma unroll
    for (int c = 0; c < 10; ++c) {
        float a = b[c];
        for (int k = 0; k < 64; ++k) a = fmaf(row[k], W[k * 10 + c], a);
        logits[c] = a;
    }
    float mx = logits[0];
#pragma unroll
    for (int c = 1; c < 10; ++c) mx = fmaxf(mx, logits[c]);
    float s = 0.0f;
#pragma unroll
    for (int c = 0; c < 10; ++c) s += expf(logits[c] - mx);
    float lse = mx + logf(s);
#pragma unroll
    for (int c = 0; c < 10; ++c) out[(size_t)g * 10 + c] = logits[c] - lse;
}

// ---------------------------------------------------------------------------
// Host launcher
// ---------------------------------------------------------------------------
static inline unsigned cdiv_u(unsigned a, unsigned b) { return (a + b - 1) / b; }

extern "C" void kernel_launch(void* const* d_in, const int* in_sizes, int n_in,
                              void* d_out, int out_size, void* d_ws, size_t ws_size,
                              hipStream_t stream) {
    const float* x      = (const float*)d_in[0];
    const int*   ei     = (const int*)d_in[1];
    const int*   batch  = (const int*)d_in[2];
    const float* W1     = (const float*)d_in[3];
    const float* b1     = (const float*)d_in[4];
    const float* W2     = (const float*)d_in[5];
    const float* b2     = (const float*)d_in[6];
    const float* W3     = (const float*)d_in[7];
    const float* b3     = (const float*)d_in[8];
    const float* lin1W  = (const float*)d_in[9];
    const float* lin1b  = (const float*)d_in[10];
    const float* lin2W  = (const float*)d_in[11];
    const float* lin2b  = (const float*)d_in[12];
    (void)n_in; (void)ws_size;

    const int N = in_sizes[0] / 64;   // 50000
    const int E = in_sizes[1] / 2;    // 800000
    const int G = out_size / 10;      // 500
    const int* src = ei;
    const int* dst = ei + E;

    // Workspace carve-up (floats)
    float* ws   = (float*)d_ws;
    size_t off  = 0;
    float* tmp  = ws + off; off += (size_t)N * 64;
    float* aggA = ws + off; off += (size_t)N * 64;
    float* aggB = ws + off; off += (size_t)N * 64;
    float* dis  = ws + off; off += (size_t)N;
    float* sums = ws + off; off += (size_t)G * 64;
    float* cnts = ws + off; off += (size_t)G;
    float* hid1 = ws + off; off += (size_t)G * 64;

    const unsigned B256 = 256;
    const unsigned NF = (unsigned)N * 64;
    const unsigned GF = (unsigned)G * 64;
    const int rowTiles = N / 16;
    const int totalTiles = rowTiles * 4;
    const unsigned gemmBlocks = cdiv_u((unsigned)totalTiles * 32, B256);
    const unsigned scatBlocks = cdiv_u((unsigned)E * 16, B256);

    // --- degree -> dis = rsqrt(1 + indeg) ---
    k_fill<<<cdiv_u(N, B256), B256, 0, stream>>>(dis, 1.0f, N);
    k_deg<<<cdiv_u(E, B256), B256, 0, stream>>>(dst, dis, E);
    k_rsqrt_inplace<<<cdiv_u(N, B256), B256, 0, stream>>>(dis, N);

    // --- layer 1: x -> aggA ---
    k_fill<<<cdiv_u(NF, B256), B256, 0, stream>>>(aggA, 0.0f, (int)NF);
    k_gemm64_wmma<<<gemmBlocks, B256, 0, stream>>>(x, W1, tmp, totalTiles);
    k_scatter<<<scatBlocks, B256, 0, stream>>>(src, dst, dis, tmp, aggA, E);
    k_finalize<<<cdiv_u(NF, B256), B256, 0, stream>>>(aggA, tmp, dis, b1, (int)NF);

    // --- layer 2: aggA -> aggB ---
    k_fill<<<cdiv_u(NF, B256), B256, 0, stream>>>(aggB, 0.0f, (int)NF);
    k_gemm64_wmma<<<gemmBlocks, B256, 0, stream>>>(aggA, W2, tmp, totalTiles);
    k_scatter<<<scatBlocks, B256, 0, stream>>>(src, dst, dis, tmp, aggB, E);
    k_finalize<<<cdiv_u(NF, B256), B256, 0, stream>>>(aggB, tmp, dis, b2, (int)NF);

    // --- layer 3: aggB -> aggA ---
    k_fill<<<cdiv_u(NF, B256), B256, 0, stream>>>(aggA, 0.0f, (int)NF);
    k_gemm64_wmma<<<gemmBlocks, B256, 0, stream>>>(aggB, W3, tmp, totalTiles);
    k_scatter<<<scatBlocks, B256, 0, stream>>>(src, dst, dis, tmp, aggA, E);
    k_finalize<<<cdiv_u(NF, B256), B256, 0, stream>>>(aggA, tmp, dis, b3, (int)NF);

    // --- global mean pool ---
    k_fill<<<cdiv_u(GF, B256), B256, 0, stream>>>(sums, 0.0f, (int)GF);
    k_fill<<<cdiv_u((unsigned)G, B256), B256, 0, stream>>>(cnts, 0.0f, G);
    k_pool_cnt<<<cdiv_u((unsigned)N, B256), B256, 0, stream>>>(batch, cnts, N);
    k_pool_sum<<<cdiv_u(NF, B256), B256, 0, stream>>>(aggA, batch, sums, (int)NF);

    // --- heads ---
    k_lin1<<<cdiv_u(GF, B256), B256, 0, stream>>>(sums, cnts, lin1W, lin1b, hid1, (int)GF);
    k_lin2_logsoftmax<<<cdiv_u((unsigned)G, B256), B256, 0, stream>>>(
        hid1, lin2W, lin2b, (float*)d_out, G);
}